// MultiHeadAttention_23854248362059
// MI455X (gfx1250) — compile-verified
//
#include <hip/hip_runtime.h>

// ---------------------------------------------------------------------------
// MI455X (gfx1250) gated multi-head attention.
//
// B=2, LQ=256, D_MODEL=512, 8 heads x 64, selector hidden 1024.
// Selector factored:  A0 = d0 @ W1[:512],  A1 = d1 @ W1[512:] + b1,
// gate(i,j) = relu(A0[i]+A1[j]) . (w2[:,1]-w2[:,0])  >  b2[0]-b2[1].
// All GEMM-shaped work uses V_WMMA_F32_16X16X4_F32 (fp32 = reference precision).
// Working set ~12 MB << 192 MB L2 -> WMMA operands fed straight from global.
// Each GEMM wave owns a 32x32 C block (2x2 WMMA tiles): per K=4 step,
// 2x b64 (A) + 4x b32 (B) feed 4 WMMAs -> 2x the FLOP/byte of 1-tile waves.
// ---------------------------------------------------------------------------

typedef __attribute__((ext_vector_type(2))) float v2f;
typedef __attribute__((ext_vector_type(8))) float v8f;

#define B_SZ 2
#define LQ 256
#define DM 512
#define NH 8
#define DKH 64
#define TWO_D 1024
#define NEGBIG -1e9f
#define LN_EPS 1e-6f

static __device__ inline v8f wmma_f32(v2f a, v2f b, v8f c) {
  // D = A(16x4) * B(4x16) + C(16x16), fp32.
  return __builtin_amdgcn_wmma_f32_16x16x4_f32(
      false, a, false, b, (short)0, c, false, false);
}

static __device__ inline float wave_sum32(float v) {
  #pragma unroll
  for (int off = 16; off >= 1; off >>= 1) v += __shfl_xor(v, off, 32);
  return v;
}

// ---------------------------------------------------------------------------
// WMMA GEMM, 32x32 C block per wave (2x2 register blocking).
// C[M,N] = scale * (A[M,K] @ Bm[K,N]) (+bias[N]) (+res[M,N]).
// Row-major, ldA=K, ldB=N, ldC=N; M,N multiples of 32, K multiple of 4.
// ---------------------------------------------------------------------------
__global__ __launch_bounds__(256) void wmma_gemm_kernel(
    const float* __restrict__ A, const float* __restrict__ Bm,
    float* __restrict__ C, int M, int N, int K, float scale,
    const float* __restrict__ bias, const float* __restrict__ residual) {
  const int tid  = threadIdx.x;
  const int wv   = tid >> 5;
  const int lane = tid & 31;
  const int hf   = lane >> 4;   // lane half selects K pair within K=4 step
  const int lr   = lane & 15;

  const int tilesN = N >> 5;
  const int total  = (M >> 5) * tilesN;
  const int id     = blockIdx.x * 8 + wv;
  if (id >= total) return;
  const int row0 = (id / tilesN) << 5;
  const int col0 = (id % tilesN) << 5;

  const float* arow0 = A + (size_t)(row0 + lr) * K;   // M-subtile 0
  const float* arow1 = arow0 + (size_t)16 * K;        // M-subtile 1
  const float* bcol0 = Bm + col0 + lr;                // N-subtile 0
  const float* bcol1 = bcol0 + 16;                    // N-subtile 1

  v8f acc00 = {0.f, 0.f, 0.f, 0.f, 0.f, 0.f, 0.f, 0.f};
  v8f acc01 = acc00, acc10 = acc00, acc11 = acc00;

  for (int k0 = 0; k0 < K; k0 += 4) {
    const int ka = k0 + 2 * hf;
    const v2f a0 = *(const v2f*)(arow0 + ka);
    const v2f a1 = *(const v2f*)(arow1 + ka);
    v2f b0, b1;
    b0.x = bcol0[(size_t)ka * N];
    b0.y = bcol0[(size_t)(ka + 1) * N];
    b1.x = bcol1[(size_t)ka * N];
    b1.y = bcol1[(size_t)(ka + 1) * N];
    acc00 = wmma_f32(a0, b0, acc00);
    acc01 = wmma_f32(a0, b1, acc01);
    acc10 = wmma_f32(a1, b0, acc10);
    acc11 = wmma_f32(a1, b1, acc11);
  }

  auto store_tile = [&](const v8f& acc, int r0, int c0) {
    #pragma unroll
    for (int r = 0; r < 8; ++r) {
      const int m = r0 + 8 * hf + r;  // VGPR r: lanes0-15 M=r, lanes16-31 M=8+r
      const int n = c0 + lr;
      float val = acc[r] * scale;
      if (bias)     val += bias[n];
      if (residual) val += residual[(size_t)m * N + n];
      C[(size_t)m * N + n] = val;
    }
  };
  store_tile(acc00, row0,      col0);
  store_tile(acc01, row0,      col0 + 16);
  store_tile(acc10, row0 + 16, col0);
  store_tile(acc11, row0 + 16, col0 + 16);
}

// ---------------------------------------------------------------------------
// Decision gate. dec[b,i,j] = mask & (dot(relu(A0[b,i]+A1[b,j]), wd) > thr)
// where wd = w2[:,1]-w2[:,0] (LDS), thr = b2[0]-b2[1]. One block per (b,i);
// each wave owns a strided set of j; 4 consecutive dims per lane (b128 loads).
// ---------------------------------------------------------------------------
__global__ __launch_bounds__(256) void decision_kernel(
    const float* __restrict__ A0, const float* __restrict__ A1,
    const float* __restrict__ w2, const float* __restrict__ b2,
    const unsigned char* __restrict__ mask, float* __restrict__ dec) {
  const int tid  = threadIdx.x;
  const int wv   = tid >> 5;
  const int lane = tid & 31;
  const int row  = blockIdx.x;          // b*LQ + i
  const int b    = row >> 8;

  __shared__ float a0row[TWO_D];
  __shared__ float wd[TWO_D];
  for (int t = tid; t < TWO_D; t += 256) {
    a0row[t] = A0[(size_t)row * TWO_D + t];
    wd[t]    = w2[2 * t + 1] - w2[2 * t];
  }
  __syncthreads();

  const float thr = b2[0] - b2[1];
  const unsigned char* mrow = mask + (size_t)row * LQ;

  for (int j = wv; j < LQ; j += 8) {
    const float* a1row = A1 + (size_t)(b * LQ + j) * TWO_D;
    float acc = 0.f;
    #pragma unroll
    for (int t = 0; t < 8; ++t) {
      const int dd = 4 * lane + 128 * t;
      const float4 a1 = *(const float4*)(a1row + dd);
      acc = fmaf(fmaxf(a0row[dd + 0] + a1.x, 0.f), wd[dd + 0], acc);
      acc = fmaf(fmaxf(a0row[dd + 1] + a1.y, 0.f), wd[dd + 1], acc);
      acc = fmaf(fmaxf(a0row[dd + 2] + a1.z, 0.f), wd[dd + 2], acc);
      acc = fmaf(fmaxf(a0row[dd + 3] + a1.w, 0.f), wd[dd + 3], acc);
    }
    acc = wave_sum32(acc);
    if (lane == 0) {
      dec[(size_t)row * LQ + j] = (mrow[j] && (acc > thr)) ? 1.0f : 0.0f;
    }
  }
}

// ---------------------------------------------------------------------------
// Attention: block = one (b,h) and one 16-row i strip.
// Phase 1: S = (Q/sqrt(dk)) @ K^T via WMMA, 2 j-tiles per wave (256 j total).
// Phase 2: gate with dec, block-wide row softmax via LDS, write attn.
// Phase 3: O = P @ V, all 8 waves: 4 n-tiles x 2 K-halves, combine via LDS.
// ---------------------------------------------------------------------------
__global__ __launch_bounds__(256) void attn_kernel(
    const float* __restrict__ Qp, const float* __restrict__ Kp,
    const float* __restrict__ Vp, const float* __restrict__ dec,
    float* __restrict__ attn, float* __restrict__ Obuf) {
  const int tid  = threadIdx.x;
  const int wv   = tid >> 5;
  const int lane = tid & 31;
  const int hf   = lane >> 4;
  const int lr   = lane & 15;

  const int itile = blockIdx.x & 15;
  const int bh    = blockIdx.x >> 4;
  const int h     = bh & 7;
  const int b     = bh >> 3;
  const int i0    = itile << 4;

  __shared__ float Pbuf[16][260];     // pad: stride%64==4 -> conflict-free
  __shared__ float red[8][16];
  __shared__ float rowmax[16];
  __shared__ float rowsum[16];
  __shared__ float Opart[4][16][16];

  const int j0a = wv * 16;
  const int j0b = (wv + 8) * 16;

  // ---- Phase 1: S tiles -------------------------------------------------
  const float* qrow  = Qp + (size_t)(b * LQ + i0 + lr) * DM + h * DKH; // pre-scaled 1/8
  const float* krowA = Kp + (size_t)(b * LQ + j0a + lr) * DM + h * DKH;
  const float* krowB = Kp + (size_t)(b * LQ + j0b + lr) * DM + h * DKH;

  v8f c0 = {0.f, 0.f, 0.f, 0.f, 0.f, 0.f, 0.f, 0.f};
  v8f c1 = c0;
  #pragma unroll
  for (int k0 = 0; k0 < DKH; k0 += 4) {
    const int ka = k0 + 2 * hf;
    const v2f a  = *(const v2f*)(qrow + ka);
    const v2f bA = *(const v2f*)(krowA + ka);
    const v2f bB = *(const v2f*)(krowB + ka);
    c0 = wmma_f32(a, bA, c0);
    c1 = wmma_f32(a, bB, c1);
  }

  // ---- Phase 2: gate + softmax ------------------------------------------
  const float* decb = dec + (size_t)b * LQ * LQ + (size_t)i0 * LQ;
  float s0[8], s1[8];
  #pragma unroll
  for (int r = 0; r < 8; ++r) {
    const int ml = r + 8 * hf;
    const float g0 = decb[ml * LQ + j0a + lr];
    const float g1 = decb[ml * LQ + j0b + lr];
    s0[r] = (g0 != 0.f) ? c0[r] : NEGBIG;
    s1[r] = (g1 != 0.f) ? c1[r] : NEGBIG;
    float pm = fmaxf(s0[r], s1[r]);
    #pragma unroll
    for (int off = 8; off >= 1; off >>= 1) pm = fmaxf(pm, __shfl_xor(pm, off, 32));
    if (lr == 0) red[wv][ml] = pm;
  }
  __syncthreads();
  if (tid < 16) {
    float m = red[0][tid];
    #pragma unroll
    for (int w = 1; w < 8; ++w) m = fmaxf(m, red[w][tid]);
    rowmax[tid] = m;
  }
  __syncthreads();

  float e0[8], e1[8];
  #pragma unroll
  for (int r = 0; r < 8; ++r) {
    const int ml = r + 8 * hf;
    const float mx = rowmax[ml];
    e0[r] = __expf(s0[r] - mx);
    e1[r] = __expf(s1[r] - mx);
    float ps = e0[r] + e1[r];
    #pragma unroll
    for (int off = 8; off >= 1; off >>= 1) ps += __shfl_xor(ps, off, 32);
    if (lr == 0) red[wv][ml] = ps;
  }
  __syncthreads();
  if (tid < 16) {
    float s = 0.f;
    #pragma unroll
    for (int w = 0; w < 8; ++w) s += red[w][tid];
    rowsum[tid] = s;
  }
  __syncthreads();

  float* attnb = attn + ((size_t)(b * NH + h) * LQ + i0) * LQ;
  #pragma unroll
  for (int r = 0; r < 8; ++r) {
    const int ml  = r + 8 * hf;
    const float inv = 1.0f / rowsum[ml];
    const float p0 = e0[r] * inv;
    const float p1 = e1[r] * inv;
    attnb[ml * LQ + j0a + lr] = p0;
    attnb[ml * LQ + j0b + lr] = p1;
    Pbuf[ml][j0a + lr] = p0;
    Pbuf[ml][j0b + lr] = p1;
  }
  __syncthreads();

  // ---- Phase 3: O = P @ V. wave w -> n-tile (w&3), K-half (w>>2). --------
  {
    const int n0    = (wv & 3) * 16;
    const int kbase = (wv >> 2) * 128;
    v8f oc = {0.f, 0.f, 0.f, 0.f, 0.f, 0.f, 0.f, 0.f};
    for (int kk = kbase; kk < kbase + 128; kk += 4) {
      const int ka = kk + 2 * hf;
      v2f a, bb;
      a.x = Pbuf[lr][ka];
      a.y = Pbuf[lr][ka + 1];
      const float* vrow = Vp + (size_t)(b * LQ + ka) * DM + h * DKH + n0 + lr;
      bb.x = vrow[0];
      bb.y = vrow[DM];
      oc = wmma_f32(a, bb, oc);
    }
    if (wv >= 4) {
      #pragma unroll
      for (int r = 0; r < 8; ++r) Opart[wv - 4][r + 8 * hf][lr] = oc[r];
    }
    __syncthreads();
    if (wv < 4) {
      #pragma unroll
      for (int r = 0; r < 8; ++r) {
        const int ml = r + 8 * hf;
        Obuf[(size_t)(b * LQ + i0 + ml) * DM + h * DKH + n0 + lr] =
            oc[r] + Opart[wv][ml][lr];
      }
    }
  }
}

// ---------------------------------------------------------------------------
// LayerNorm over last dim (512), one block (256 threads) per row.
// ---------------------------------------------------------------------------
__global__ __launch_bounds__(256) void ln_kernel(
    const float* __restrict__ x, const float* __restrict__ g,
    const float* __restrict__ bta, float* __restrict__ out) {
  const int tid  = threadIdx.x;
  const int wv   = tid >> 5;
  const int lane = tid & 31;
  const size_t row = blockIdx.x;

  __shared__ float sbuf[8];
  __shared__ float smean, svar;

  const float v0 = x[row * DM + tid];
  const float v1 = x[row * DM + tid + 256];

  float s = wave_sum32(v0 + v1);
  if (lane == 0) sbuf[wv] = s;
  __syncthreads();
  if (tid == 0) {
    float t = 0.f;
    #pragma unroll
    for (int w = 0; w < 8; ++w) t += sbuf[w];
    smean = t * (1.0f / DM);
  }
  __syncthreads();
  const float mu = smean;
  const float d0 = v0 - mu, d1 = v1 - mu;
  float s2 = wave_sum32(d0 * d0 + d1 * d1);
  if (lane == 0) sbuf[wv] = s2;
  __syncthreads();
  if (tid == 0) {
    float t = 0.f;
    #pragma unroll
    for (int w = 0; w < 8; ++w) t += sbuf[w];
    svar = t * (1.0f / DM);
  }
  __syncthreads();
  const float inv = rsqrtf(svar + LN_EPS);
  out[row * DM + tid]       = d0 * inv * g[tid]       + bta[tid];
  out[row * DM + tid + 256] = d1 * inv * g[tid + 256] + bta[tid + 256];
}

// ---------------------------------------------------------------------------
extern "C" void kernel_launch(void* const* d_in, const int* in_sizes, int n_in,
                              void* d_out, int out_size, void* d_ws, size_t ws_size,
                              hipStream_t stream) {
  (void)in_sizes; (void)n_in; (void)out_size; (void)ws_size;

  const float* q     = (const float*)d_in[0];
  const float* k     = (const float*)d_in[1];
  const float* v     = (const float*)d_in[2];
  const unsigned char* mask = (const unsigned char*)d_in[3];
  const float* w_qs  = (const float*)d_in[4];
  const float* w_ks  = (const float*)d_in[5];
  const float* w_vs  = (const float*)d_in[6];
  const float* w_ds0 = (const float*)d_in[7];
  const float* w_ds1 = (const float*)d_in[8];
  const float* dc_w1 = (const float*)d_in[9];
  const float* dc_b1 = (const float*)d_in[10];
  const float* dc_w2 = (const float*)d_in[11];
  const float* dc_b2 = (const float*)d_in[12];
  const float* fc_w  = (const float*)d_in[13];
  const float* ln_g  = (const float*)d_in[14];
  const float* ln_b  = (const float*)d_in[15];

  // workspace layout (floats)
  float* ws  = (float*)d_ws;
  const size_t MAT = (size_t)B_SZ * LQ * DM;      // 262144
  float* Qp  = ws;                                 // Q proj, pre-scaled 1/8
  float* Kp  = Qp + MAT;
  float* Vp  = Kp + MAT;
  float* d0b = Vp + MAT;
  float* d1b = d0b + MAT;
  float* A0  = d1b + MAT;                          // [512,1024]
  float* A1  = A0 + (size_t)B_SZ * LQ * TWO_D;     // [512,1024] (+b1)
  float* Ob  = A1 + (size_t)B_SZ * LQ * TWO_D;     // attention output, pre-fc
  float* tmp = Ob + MAT;                           // fc output + residual

  // output layout: out | attn | dec
  float* out_p  = (float*)d_out;
  float* attn_p = out_p + MAT;
  float* dec_p  = attn_p + (size_t)B_SZ * NH * LQ * LQ;

  const dim3 blk(256);
  const int M = B_SZ * LQ;   // 512
  const float invT = 0.125f; // 1/sqrt(64)

  // 32x32 wave blocks: grid = (M/32)*(N/32)/8 waves per block
  // proj GEMMs: 16*16 = 256 wave blocks -> 32 CTAs
  wmma_gemm_kernel<<<32, blk, 0, stream>>>(q, w_qs, Qp, M, DM, DM, invT, nullptr, nullptr);
  wmma_gemm_kernel<<<32, blk, 0, stream>>>(k, w_ks, Kp, M, DM, DM, 1.f, nullptr, nullptr);
  wmma_gemm_kernel<<<32, blk, 0, stream>>>(v, w_vs, Vp, M, DM, DM, 1.f, nullptr, nullptr);
  wmma_gemm_kernel<<<32, blk, 0, stream>>>(q, w_ds0, d0b, M, DM, DM, 1.f, nullptr, nullptr);
  wmma_gemm_kernel<<<32, blk, 0, stream>>>(v, w_ds1, d1b, M, DM, DM, 1.f, nullptr, nullptr);

  // selector factors: 16*32 = 512 wave blocks -> 64 CTAs
  wmma_gemm_kernel<<<64, blk, 0, stream>>>(d0b, dc_w1, A0, M, TWO_D, DM, 1.f, nullptr, nullptr);
  wmma_gemm_kernel<<<64, blk, 0, stream>>>(d1b, dc_w1 + (size_t)DM * TWO_D, A1, M, TWO_D, DM,
                                           1.f, dc_b1, nullptr);

  // pairwise gate -> dec output
  decision_kernel<<<M, blk, 0, stream>>>(A0, A1, dc_w2, dc_b2, mask, dec_p);

  // attention (B*NH*(LQ/16) = 256 blocks) -> attn output + Ob
  attn_kernel<<<256, blk, 0, stream>>>(Qp, Kp, Vp, dec_p, attn_p, Ob);

  // output projection + residual, then layernorm
  wmma_gemm_kernel<<<32, blk, 0, stream>>>(Ob, fc_w, tmp, M, DM, DM, 1.f, nullptr, q);
  ln_kernel<<<M, blk, 0, stream>>>(tmp, ln_g, ln_b, out_p);
}